// ClassTokenNested_50251117363827
// MI455X (gfx1250) — compile-verified
//
#include <hip/hip_runtime.h>

typedef float v4f __attribute__((ext_vector_type(4)));

// ---------------------------------------------------------------------------
// Kernel 1: token scatter + class-row boundary detection.
//   out[i + seg[i] + 1] = x[i]            (row-wise, D floats, NT b128 path)
//   clsRow[s] = offsets[s] + s            (found via seg boundary detection)
//
// 64 threads per row, 4 rows per 256-thread block.  The ~516MB stream exceeds
// the 192MB L2 and is never re-read, so TH=NT on both load and store avoids
// cache thrash.  Boundary detection: if seg[row-1] < seg[row], then for every
// s in (seg[row-1], seg[row]], offsets[s] == row, class row == row + s.  The
// last row group also emits trailing empty sequences (offsets[s] == T).
// This replaces a ~18-deep dependent-load binary search per sequence with
// work kernel 1 gets for free while streaming seg_ids.
// ---------------------------------------------------------------------------
__global__ __launch_bounds__(256) void ctn_tok_scatter(
    const float* __restrict__ x, const int* __restrict__ seg,
    float* __restrict__ out, int* __restrict__ clsRow,
    int T, int B, int D4 /* D/4 */) {
  const int row = blockIdx.x * 4 + (threadIdx.x >> 6);
  const int c0  = threadIdx.x & 63;
  if (row >= T) return;

  const int s = seg[row];                 // uniform across the 64 row-threads

  // Lane 0 of each row group emits class-row indices for boundaries at `row`.
  // Rare (B total hits across T rows) -> negligible divergence.
  if (c0 == 0 && clsRow != nullptr) {
    const int a = (row == 0) ? -1 : seg[row - 1];
    for (int q = a + 1; q <= s; ++q)      // offsets[q] == row
      clsRow[q] = row + q;
    if (row == T - 1) {
      for (int q = s + 1; q < B; ++q)     // trailing empty seqs: offsets == T
        clsRow[q] = T + q;
    }
  }

  const long long srcRow = (long long)row * D4;
  const long long dstRow = (long long)(row + s + 1) * D4;

  const v4f* __restrict__ xv = (const v4f*)x;
  v4f* __restrict__ ov = (v4f*)out;

  // D==256 -> exactly one iteration; loop kept for generality.
  for (int c = c0; c < D4; c += 64) {
    v4f v = __builtin_nontemporal_load(xv + srcRow + c);   // global_load_b128 th:NT
    __builtin_nontemporal_store(v, ov + dstRow + c);       // global_store_b128 th:NT
  }
}

// ---------------------------------------------------------------------------
// Kernel 2: class-token broadcast.  out[clsRow[s]] = weight
// 64 threads per sequence, 4 sequences per 256-thread block.  With clsRow
// from kernel 1 this is a trivial 8KB gather + 2MB write.  If no workspace is
// available (clsRow == nullptr), falls back to a wave-uniform binary search.
// ---------------------------------------------------------------------------
__global__ __launch_bounds__(256) void ctn_cls_scatter(
    const float* __restrict__ w, const int* __restrict__ seg,
    const int* __restrict__ clsRow, float* __restrict__ out,
    int T, int B, int D4 /* D/4 */) {
  const int s  = blockIdx.x * 4 + (threadIdx.x >> 6);
  const int c0 = threadIdx.x & 63;
  if (s >= B) return;

  int r;
  if (clsRow != nullptr) {
    r = clsRow[s];
  } else {
    // offsets[s] = first index i with seg[i] >= s (searchsorted side='left')
    int lo = 0, hi = T;
    while (lo < hi) {
      int mid = (lo + hi) >> 1;
      if (seg[mid] < s) lo = mid + 1; else hi = mid;
    }
    r = lo + s;
  }

  const long long dstRow = (long long)r * D4;
  const v4f* __restrict__ wv = (const v4f*)w;
  v4f* __restrict__ ov = (v4f*)out;

  for (int c = c0; c < D4; c += 64) {
    v4f v = wv[c];                                          // tiny, stays cached
    __builtin_nontemporal_store(v, ov + dstRow + c);        // global_store_b128 th:NT
  }
}

// ---------------------------------------------------------------------------
// Host launcher.
// Inputs (setup_inputs order):
//   d_in[0] = x_flat  [T, D] float32      (in_sizes[0] = T*D)
//   d_in[1] = weight  [1, D] float32      (in_sizes[1] = D)
//   d_in[2] = seg_ids [T]    int          (in_sizes[2] = T)
//   d_in[3] = num_segments   int scalar   (in_sizes[3] = 1)  -- device only
// Output: out [(T+B), D] float32, out_size = (T+B)*D  =>  B = out_size/D - T
// Token rows + class rows exactly partition the output, so no zero-fill pass.
// clsRow table (B ints, 8KB) lives in d_ws; fully rewritten every call, so
// graph replays are self-contained.
// ---------------------------------------------------------------------------
extern "C" void kernel_launch(void* const* d_in, const int* in_sizes, int n_in,
                              void* d_out, int out_size, void* d_ws, size_t ws_size,
                              hipStream_t stream) {
  const float* x   = (const float*)d_in[0];
  const float* w   = (const float*)d_in[1];
  const int*   seg = (const int*)d_in[2];
  float*       out = (float*)d_out;

  const int D  = in_sizes[1];        // 256
  const int T  = in_sizes[2];        // 262144
  const int B  = out_size / D - T;   // 2048 (num_segments, derived host-side)
  const int D4 = D >> 2;             // 64 float4 per row

  int* clsRow = (ws_size >= (size_t)B * sizeof(int)) ? (int*)d_ws : nullptr;

  const int rowsPerBlock = 4;
  dim3 blk(256);

  dim3 grid1((T + rowsPerBlock - 1) / rowsPerBlock);
  ctn_tok_scatter<<<grid1, blk, 0, stream>>>(x, seg, out, clsRow, T, B, D4);

  dim3 grid2((B + rowsPerBlock - 1) / rowsPerBlock);
  ctn_cls_scatter<<<grid2, blk, 0, stream>>>(w, seg, clsRow, out, T, B, D4);
}